// UnsupervisedSegmentation_81123342287548
// MI455X (gfx1250) — compile-verified
//
#include <hip/hip_runtime.h>
#include <math.h>

// ---- problem constants (from reference setup_inputs) ----
#define B_DIM   64
#define S_DIM   131072
#define WIN     20
#define HALF    10
#define CHUNK   4096
#define NCHUNK  (S_DIM / CHUNK)          // 32
#define STAGE_N (CHUNK + WIN)            // 4116 floats staged per block
#define BLOCK   256                      // 8 wave32
#define TILES_PER_WAVE 2                 // 8 waves * 2 tiles * 256 outputs = 4096

typedef __attribute__((ext_vector_type(2))) float     v2f;
typedef __attribute__((ext_vector_type(4))) float     f4;
typedef __attribute__((ext_vector_type(8))) float     v8f;
typedef __attribute__((ext_vector_type(4))) unsigned  u32x4;
typedef __attribute__((ext_vector_type(4))) int       i32x4;
typedef __attribute__((ext_vector_type(8))) int       i32x8;

#if defined(__has_builtin)
#  if __has_builtin(__builtin_amdgcn_tensor_load_to_lds) && __has_builtin(__builtin_amdgcn_s_wait_tensorcnt)
#    define HAVE_TDM 1
#  endif
#  if __has_builtin(__builtin_amdgcn_wmma_f32_16x16x4_f32)
#    define HAVE_WMMA4 1
#  endif
#endif

// ---- monotone float<->uint encode so integer atomicMin/Max give float min/max ----
__device__ __forceinline__ unsigned enc_f32(float f) {
    unsigned u = __float_as_uint(f);
    return (u & 0x80000000u) ? ~u : (u | 0x80000000u);
}
__device__ __forceinline__ float dec_f32(unsigned e) {
    unsigned u = (e & 0x80000000u) ? (e ^ 0x80000000u) : ~e;
    return __uint_as_float(u);
}

// ---- stage one chunk (+halo, reflect-padded) of one signal row into LDS ----
// Interior chunks: single TDM descriptor moves STAGE_N contiguous floats to LDS
// (issued by wave 0 only; TDM ignores EXEC and issues once per wave).
// Edge chunks (2 of 32 per row): cooperative loads with reflect index mapping.
__device__ __forceinline__ void stage_row(float* sh, const float* rowp, int c0) {
#if defined(HAVE_TDM)
    if (c0 >= HALF && (c0 + CHUNK + HALF) <= S_DIM) {
        if (threadIdx.x < 32) {   // wave 0 only -> exactly one DMA per block
            unsigned long long ga =
                (unsigned long long)(const void*)(rowp + (c0 - HALF));
            unsigned lds_byte = (unsigned)(unsigned long long)(const void*)sh;

            u32x4 g0;
            g0[0] = 1u;                                   // count=1, no gather
            g0[1] = lds_byte;                             // lds_addr (bytes)
            g0[2] = (unsigned)(ga & 0xFFFFFFFFu);         // global_addr[31:0]
            g0[3] = (unsigned)((ga >> 32) & 0x1FFFFFFu)   // global_addr[56:32]
                    | (2u << 30);                         // type=2 (image)

            const unsigned dim0 = (unsigned)STAGE_N;      // elements (4B each)
            i32x8 g1;
            g1[0] = (int)(2u << 16);                      // data_size=2 (4 bytes)
            g1[1] = (int)((dim0 & 0xFFFFu) << 16);        // tensor_dim0[15:0]
            g1[2] = (int)(((dim0 >> 16) & 0xFFFFu)        // tensor_dim0[31:16]
                    | (1u << 16));                        // tensor_dim1 = 1
            g1[3] = (int)((dim0 & 0xFFFFu) << 16);        // tile_dim0 = STAGE_N
            g1[4] = 1;                                    // tile_dim1=1, tile_dim2=0
            g1[5] = (int)dim0;                            // tensor_dim0_stride lo
            g1[6] = 0;
            g1[7] = 0;
            i32x4 z4; z4[0] = 0; z4[1] = 0; z4[2] = 0; z4[3] = 0;
            i32x8 z8;
#pragma unroll
            for (int i = 0; i < 8; ++i) z8[i] = 0;

            // 6-arg form (upstream clang-23 / therock-10.0 signature)
            __builtin_amdgcn_tensor_load_to_lds(g0, g1, z4, z4, z8, 0);
            __builtin_amdgcn_s_wait_tensorcnt(0);
        }
        return;   // caller barriers before use
    }
#endif
    for (int t = (int)threadIdx.x; t < STAGE_N; t += BLOCK) {
        int g = c0 - HALF + t;
        g = (g < 0) ? -g : g;                       // reflect (no edge repeat)
        g = (g >= S_DIM) ? (2 * S_DIM - 2 - g) : g;
        sh[t] = rowp[g];
    }
}

// ---- 256-output tile: window sum & sum-of-squares via banded-ones WMMA ----
// D[m,n] = sum_j W[m,j] * x[L0 + 16n + j],  W[m,j]=1 iff m <= j <= m+19, j in [0,35].
// 9 chained V_WMMA_F32_16X16X4_F32 per statistic. D layout: lane holds N=lane&15,
// vgpr r holds M=r+(lane>=16?8:0) -> each lane owns 8 consecutive output positions.
__device__ __forceinline__ void tile_sums(const float* sh, int L0, int lane,
                                          float* s8, float* q8) {
    const int nm   = lane & 15;            // = M for A-matrix, = N for B-matrix
    const int koff = (lane >= 16) ? 2 : 0; // K sub-column for this half-wave
#if defined(HAVE_WMMA4)
    v8f s = {0.f,0.f,0.f,0.f,0.f,0.f,0.f,0.f};
    v8f q = {0.f,0.f,0.f,0.f,0.f,0.f,0.f,0.f};
#pragma unroll
    for (int k = 0; k < 9; ++k) {
        const int j0 = 4 * k + koff;
        v2f a, b, b2;
        a[0] = (j0     >= nm && j0     <= nm + (WIN - 1)) ? 1.0f : 0.0f;
        a[1] = (j0 + 1 >= nm && j0 + 1 <= nm + (WIN - 1)) ? 1.0f : 0.0f;
        const float x0 = sh[L0 + nm * 16 + j0];
        const float x1 = sh[L0 + nm * 16 + j0 + 1];
        b[0]  = x0;       b[1]  = x1;
        b2[0] = x0 * x0;  b2[1] = x1 * x1;
        s = __builtin_amdgcn_wmma_f32_16x16x4_f32(false, a, false, b,  (short)0, s, false, false);
        q = __builtin_amdgcn_wmma_f32_16x16x4_f32(false, a, false, b2, (short)0, q, false, false);
    }
#pragma unroll
    for (int r = 0; r < 8; ++r) { s8[r] = s[r]; q8[r] = q[r]; }
#else
    const int mbase = (lane >= 16) ? 8 : 0;
#pragma unroll
    for (int r = 0; r < 8; ++r) {
        const int base = L0 + nm * 16 + mbase + r;
        float ss = 0.f, qq = 0.f;
#pragma unroll
        for (int t = 0; t < WIN; ++t) {
            const float x = sh[base + t];
            ss += x; qq += x * x;
        }
        s8[r] = ss; q8[r] = qq;
    }
#endif
}

__device__ __forceinline__ float local_var_of(float s, float q) {
    return (q - s * s * (1.0f / WIN)) * (1.0f / (WIN - 1));
}

// ---- kernel 0: reset per-row min/max accumulators in workspace ----
__global__ void useg_init(unsigned* __restrict__ mm) {
    const int t = threadIdx.x;
    if (t < B_DIM)            mm[t] = 0xFFFFFFFFu;          // encoded +inf (min slot)
    else if (t < 2 * B_DIM)   mm[t] = 0u;                   // encoded -inf (max slot)
}

// ---- kernel 1: local variance + per-row min/max ----
__global__ void useg_pass1(const float* __restrict__ sig, unsigned* __restrict__ mm) {
    __shared__ float sh[STAGE_N + 12];
    __shared__ unsigned rmin[BLOCK / 32], rmax[BLOCK / 32];

    const int c0 = (int)blockIdx.x * CHUNK;
    const int b  = (int)blockIdx.y;
    const float* rowp = sig + (long long)b * S_DIM;

    stage_row(sh, rowp, c0);
    __syncthreads();

    const int lane = (int)threadIdx.x & 31;
    const int wave = (int)threadIdx.x >> 5;

    float fmn =  3.4028235e38f;
    float fmx = -3.4028235e38f;
#pragma unroll
    for (int rep = 0; rep < TILES_PER_WAVE; ++rep) {
        const int L0 = (wave * TILES_PER_WAVE + rep) * 256;
        float s8[8], q8[8];
        tile_sums(sh, L0, lane, s8, q8);
#pragma unroll
        for (int r = 0; r < 8; ++r) {
            const float lv = local_var_of(s8[r], q8[r]);
            fmn = fminf(fmn, lv);
            fmx = fmaxf(fmx, lv);
        }
    }
#pragma unroll
    for (int off = 16; off > 0; off >>= 1) {
        fmn = fminf(fmn, __shfl_xor(fmn, off, 32));
        fmx = fmaxf(fmx, __shfl_xor(fmx, off, 32));
    }
    if (lane == 0) { rmin[wave] = enc_f32(fmn); rmax[wave] = enc_f32(fmx); }
    __syncthreads();
    if (threadIdx.x == 0) {
        unsigned mn = 0xFFFFFFFFu, mx = 0u;
#pragma unroll
        for (int w = 0; w < BLOCK / 32; ++w) {
            mn = (rmin[w] < mn) ? rmin[w] : mn;
            mx = (rmax[w] > mx) ? rmax[w] : mx;
        }
        atomicMin(&mm[b], mn);
        atomicMax(&mm[B_DIM + b], mx);
    }
}

// ---- kernel 2: normalize + emit logits/probs (non-temporal b128 stores) ----
__global__ void useg_pass2(const float* __restrict__ sig,
                           const unsigned* __restrict__ mm,
                           float* __restrict__ out) {
    __shared__ float sh[STAGE_N + 12];

    const int c0 = (int)blockIdx.x * CHUNK;
    const int b  = (int)blockIdx.y;
    const float* rowp = sig + (long long)b * S_DIM;

    stage_row(sh, rowp, c0);
    __syncthreads();

    const float vmin = dec_f32(mm[b]);            // uniform -> scalar loads
    const float vmax = dec_f32(mm[B_DIM + b]);
    const float inv  = 1.0f / (vmax - vmin + 1e-8f);

    const int lane  = (int)threadIdx.x & 31;
    const int wave  = (int)threadIdx.x >> 5;
    const int nm    = lane & 15;
    const int mbase = (lane >= 16) ? 8 : 0;

    const size_t half = (size_t)B_DIM * S_DIM * 2;   // logits | probs
    float* logits = out;
    float* probs  = out + half;

#pragma unroll
    for (int rep = 0; rep < TILES_PER_WAVE; ++rep) {
        const int L0 = (wave * TILES_PER_WAVE + rep) * 256;
        float s8[8], q8[8];
        tile_sums(sh, L0, lane, s8, q8);

        float p0[8], p1[8];
#pragma unroll
        for (int r = 0; r < 8; ++r) {
            const float lv = local_var_of(s8[r], q8[r]);
            const float nv = (lv - vmin) * inv;
            p0[r] = 1.0f - nv;
            p1[r] = nv;
        }

        const int    i0  = c0 + L0 + nm * 16 + mbase;     // 8 consecutive positions
        const size_t off = ((size_t)b * S_DIM + (size_t)i0) * 2;  // 64B aligned
        f4* lp = (f4*)(logits + off);
        f4* pp = (f4*)(probs + off);
#pragma unroll
        for (int pr = 0; pr < 4; ++pr) {
            const int r = 2 * pr;
            f4 pv, lv4;
            pv[0] = p0[r];     pv[1] = p1[r];
            pv[2] = p0[r + 1]; pv[3] = p1[r + 1];
            lv4[0] = __logf(pv[0] + 1e-10f);
            lv4[1] = __logf(pv[1] + 1e-10f);
            lv4[2] = __logf(pv[2] + 1e-10f);
            lv4[3] = __logf(pv[3] + 1e-10f);
            __builtin_nontemporal_store(lv4, lp + pr);   // 134 MB write-once: NT
            __builtin_nontemporal_store(pv,  pp + pr);
        }
    }
}

extern "C" void kernel_launch(void* const* d_in, const int* in_sizes, int n_in,
                              void* d_out, int out_size, void* d_ws, size_t ws_size,
                              hipStream_t stream) {
    (void)in_sizes; (void)n_in; (void)out_size; (void)ws_size;
    // d_in[0] = context (64x131072x16) -- unused by the reference; never touch it.
    const float* sig = (const float*)d_in[1];
    float*       out = (float*)d_out;
    unsigned*    mm  = (unsigned*)d_ws;   // [0,64): row-min enc, [64,128): row-max enc

    hipLaunchKernelGGL(useg_init, dim3(1), dim3(128), 0, stream, mm);
    dim3 grid(NCHUNK, B_DIM);
    hipLaunchKernelGGL(useg_pass1, grid, dim3(BLOCK), 0, stream, sig, mm);
    hipLaunchKernelGGL(useg_pass2, grid, dim3(BLOCK), 0, stream, sig, mm, out);
}